// CategoricalProjection_9852654977713
// MI455X (gfx1250) — compile-verified
//
#include <hip/hip_runtime.h>
#include <stdint.h>

#ifndef __has_builtin
#define __has_builtin(x) 0
#endif

// ---------------------------------------------------------------------------
// C51 categorical projection, MI455X (gfx1250).
//  * Memory-bound: ~218 MB traffic -> ~9.4 us floor at 23.3 TB/s. No matmul
//    structure -> WMMA not applicable; the CDNA5 win is the async global->LDS
//    data path (coalesced 16B/lane transfers for the 204B-per-row prob tile).
//  * One thread per row; private 53-float LDS accumulator per lane
//    (odd stride -> bank-conflict-free across a wave32); ds_add_f32 atomics
//    for the two-bin scatter; coalesced nontemporal writeout.
// ---------------------------------------------------------------------------

namespace {
constexpr int   kAtoms        = 51;
constexpr float kGamma        = 0.99f;
constexpr float kInvDelta     = 2.5f;   // 1 / ((10 - (-10)) / 50)
constexpr int   kTpb          = 256;
constexpr int   kRowsPerBlock = 256;
constexpr int   kAccStride    = 53;     // 51 bins + absorber slot; odd stride
constexpr int   kTileFloats   = kRowsPerBlock * kAtoms;      // 13056
constexpr int   kAccFloats    = kRowsPerBlock * kAccStride;  // 13568
}

#define GLOBAL_AS __attribute__((address_space(1)))
#define LDS_AS    __attribute__((address_space(3)))

typedef int v4i __attribute__((vector_size(4 * sizeof(int))));

__device__ __forceinline__ void async_g2l_b128(const void* g, void* l) {
#if __has_builtin(__builtin_amdgcn_global_load_async_to_lds_b128)
  __builtin_amdgcn_global_load_async_to_lds_b128(
      (GLOBAL_AS v4i*)g, (LDS_AS v4i*)l, 0, 0);
#else
  // Low 32 bits of the flat address of a __shared__ object are the LDS byte
  // offset (ISA 10.2 aperture decode).
  unsigned lofs = (unsigned)(size_t)l;
  asm volatile("global_load_async_to_lds_b128 %0, %1, off"
               :: "v"(lofs), "v"(g) : "memory");
#endif
}

__device__ __forceinline__ void async_g2l_b32(const void* g, void* l) {
#if __has_builtin(__builtin_amdgcn_global_load_async_to_lds_b32)
  __builtin_amdgcn_global_load_async_to_lds_b32(
      (GLOBAL_AS int*)g, (LDS_AS int*)l, 0, 0);
#else
  unsigned lofs = (unsigned)(size_t)l;
  asm volatile("global_load_async_to_lds_b32 %0, %1, off"
               :: "v"(lofs), "v"(g) : "memory");
#endif
}

__device__ __forceinline__ void wait_async0() {
#if __has_builtin(__builtin_amdgcn_s_wait_asynccnt)
  __builtin_amdgcn_s_wait_asynccnt(0);
#else
  asm volatile("s_wait_asynccnt 0x0" ::: "memory");
#endif
}

__global__ __launch_bounds__(kTpb) void c51_project_kernel(
    const float* __restrict__ reward,
    const float* __restrict__ probs,
    const float* __restrict__ not_done,
    float* __restrict__ out,
    int bs)
{
  __shared__ __align__(16) float pbuf[kTileFloats];  // 52224 B staged prob tile
  __shared__ __align__(16) float acc[kAccFloats];    // 54272 B accumulators

  const int    tid  = threadIdx.x;
  const size_t row0 = (size_t)blockIdx.x * kRowsPerBlock;

  if (row0 + kRowsPerBlock <= (size_t)bs) {
    // ---------------- fast path: full 256-row tile ----------------
    // Coalesced async copy: 13056 floats = 12 x (256 lanes * 16B) + 3 x (256 * 4B)
    const char* gsrc = (const char*)(probs + row0 * kAtoms);
    char*       ldst = (char*)pbuf;
#pragma unroll
    for (int rnd = 0; rnd < 12; ++rnd) {
      const int off = (rnd * kTpb + tid) * 16;
      async_g2l_b128(gsrc + off, ldst + off);
    }
#pragma unroll
    for (int rnd = 0; rnd < 3; ++rnd) {
      const int off = 12 * kTpb * 16 + (rnd * kTpb + tid) * 4;
      async_g2l_b32(gsrc + off, ldst + off);
    }

    // Overlap with the async copy: scalar row inputs + accumulator zeroing.
    const float r  = reward[row0 + tid];
    const float nd = not_done[row0 + tid];
    for (int i = tid; i < kAccFloats; i += kTpb) acc[i] = 0.0f;

    wait_async0();
    __syncthreads();

    // idx(a) = clamp((r + 0.99*nd*(-10 + 0.4a) + 10) * 2.5, 0, 50)
    //        = clamp(c0 + slope*a, 0, 50)   with slope = 0.99*nd.
    const float slope = kGamma * nd;
    const float c0    = fmaf(kInvDelta, r, 25.0f) - 25.0f * slope;

    const float* myP   = &pbuf[tid * kAtoms];     // stride 51 (odd) -> no conflicts
    float*       myAcc = &acc[tid * kAccStride];  // stride 53 (odd) -> no conflicts

#pragma unroll
    for (int a = 0; a < kAtoms; ++a) {
      float idx = fmaf(slope, (float)a, c0);
      idx = fminf(fmaxf(idx, 0.0f), 50.0f);
      const float lo = floorf(idx);
      const float f  = idx - lo;          // == 0 exactly in the "same bin" case
      const int   li = (int)lo;           // 0..50; li+1 <= 51 (absorber slot)
      const float pa = myP[a];
      const float hi_add = f * pa;
      const float lo_add = pa - hi_add;   // (1-f)*p; == p when f == 0
      atomicAdd(&myAcc[li],     lo_add);  // ds_add_f32, lane-private region
      atomicAdd(&myAcc[li + 1], hi_add);
    }

    __syncthreads();

    // Coalesced, nontemporal writeout (output is never re-read by us).
    float* gout = out + row0 * kAtoms;
#pragma unroll
    for (int i = tid; i < kTileFloats; i += kTpb) {
      const int rr = i / kAtoms;
      const int k  = i - rr * kAtoms;
      __builtin_nontemporal_store(acc[rr * kAccStride + k], &gout[i]);
    }
  } else {
    // ---------------- tail path: bs not a multiple of 256 ----------------
    const size_t row   = row0 + tid;
    float*       myAcc = &acc[tid * kAccStride];
#pragma unroll
    for (int i = 0; i < kAccStride; ++i) myAcc[i] = 0.0f;
    if (row < (size_t)bs) {
      const float r     = reward[row];
      const float slope = kGamma * not_done[row];
      const float c0    = fmaf(kInvDelta, r, 25.0f) - 25.0f * slope;
      const float* gP   = probs + row * kAtoms;
#pragma unroll
      for (int a = 0; a < kAtoms; ++a) {
        float idx = fminf(fmaxf(fmaf(slope, (float)a, c0), 0.0f), 50.0f);
        const float lo = floorf(idx);
        const float f  = idx - lo;
        const int   li = (int)lo;
        const float pa = gP[a];
        const float hi_add = f * pa;
        atomicAdd(&myAcc[li],     pa - hi_add);
        atomicAdd(&myAcc[li + 1], hi_add);
      }
      float* gO = out + row * kAtoms;
#pragma unroll
      for (int k = 0; k < kAtoms; ++k) gO[k] = myAcc[k];
    }
  }
}

extern "C" void kernel_launch(void* const* d_in, const int* in_sizes, int n_in,
                              void* d_out, int out_size, void* d_ws, size_t ws_size,
                              hipStream_t stream) {
  (void)n_in; (void)out_size; (void)d_ws; (void)ws_size;
  const float* reward   = (const float*)d_in[0];
  const float* probs    = (const float*)d_in[1];
  const float* not_done = (const float*)d_in[2];
  float*       out      = (float*)d_out;

  const int bs     = in_sizes[0];                       // reward is (BS, 1)
  const int blocks = (bs + kRowsPerBlock - 1) / kRowsPerBlock;  // 2048 for BS=524288

  hipLaunchKernelGGL(c51_project_kernel, dim3(blocks), dim3(kTpb), 0, stream,
                     reward, probs, not_done, out, bs);
}